// ModelNew_36215164240294
// MI455X (gfx1250) — compile-verified
//
#include <hip/hip_runtime.h>

typedef __attribute__((ext_vector_type(2))) float v2f;
typedef __attribute__((ext_vector_type(8))) float v8f;

#define NF 8192                     // IN_FEATURES == OUT_FEATURES == 8192
#define ROW_CHUNKS 16
#define ROWS_PER_CHUNK (NF / ROW_CHUNKS)   // 512

// ---------------------------------------------------------------------------
// Kernel A: column sums of weight via WMMA.
// D(16x16) = ones(16x4) @ Wtile(4x16) + C  accumulated over 512 rows per wave.
// Each wave owns a 16-column strip; lanes 0-15 of D VGPR0 hold the 16 sums.
// ---------------------------------------------------------------------------
__global__ void __launch_bounds__(256)
colsum_wmma_kernel(const float* __restrict__ W, float* __restrict__ partial) {
  const int lane    = threadIdx.x & 31;
  const int wave    = threadIdx.x >> 5;
  const int laneLow = lane & 15;
  const int laneHi  = lane >> 4;                 // 0 or 1
  const int c0      = blockIdx.x * 128 + wave * 16;
  const int r0      = blockIdx.y * ROWS_PER_CHUNK;

  v2f a; a[0] = 1.0f; a[1] = 1.0f;               // all-ones A matrix
  v8f acc0 = {};
  v8f acc1 = {};

  // lane reads column (c0+laneLow); VGPR0 covers rows {k+laneHi},
  // VGPR1 covers rows {k+2+laneHi}.  (A==1 makes the K mapping immaterial.)
  const float* p = W + (size_t)(r0 + laneHi) * NF + (size_t)(c0 + laneLow);

  #pragma unroll 4
  for (int k = 0; k < ROWS_PER_CHUNK; k += 8) {
    v2f b0, b1;
    b0[0] = p[0];
    b0[1] = p[2 * (size_t)NF];
    b1[0] = p[4 * (size_t)NF];
    b1[1] = p[6 * (size_t)NF];
    acc0 = __builtin_amdgcn_wmma_f32_16x16x4_f32(false, a, false, b0,
                                                 (short)0, acc0, false, false);
    acc1 = __builtin_amdgcn_wmma_f32_16x16x4_f32(false, a, false, b1,
                                                 (short)0, acc1, false, false);
    p += 8 * (size_t)NF;
  }

  v8f c = acc0 + acc1;
  if (lane < 16) {
    partial[(size_t)blockIdx.y * NF + (size_t)(c0 + lane)] = c[0];
  }
}

// ---------------------------------------------------------------------------
// Kernel B: fold the 16 row-chunk partials per column.
// ---------------------------------------------------------------------------
__global__ void __launch_bounds__(256)
reduce_partials_kernel(const float* __restrict__ partial,
                       float* __restrict__ wsum) {
  const int i = blockIdx.x * 256 + threadIdx.x;
  float s = 0.0f;
  #pragma unroll
  for (int t = 0; t < ROW_CHUNKS; ++t) s += partial[(size_t)t * NF + i];
  wsum[i] = s;
}

// ---------------------------------------------------------------------------
// Kernel C: sum of bias (single block, deterministic LDS tree reduce).
// ---------------------------------------------------------------------------
__global__ void __launch_bounds__(256)
bias_reduce_kernel(const float* __restrict__ bias, float* __restrict__ bsum) {
  __shared__ float sm[256];
  float s = 0.0f;
  for (int i = threadIdx.x; i < NF; i += 256) s += bias[i];
  sm[threadIdx.x] = s;
  __syncthreads();
  for (int st = 128; st > 0; st >>= 1) {
    if (threadIdx.x < st) sm[threadIdx.x] += sm[threadIdx.x + st];
    __syncthreads();
  }
  if (threadIdx.x == 0) bsum[0] = sm[0];
}

// ---------------------------------------------------------------------------
// Kernel D: GEMV  out[b] = dot(x[b,:], wsum) + bias_sum.
// One 256-thread block per batch row; float4 streaming loads; wsum stays in L2.
// ---------------------------------------------------------------------------
__global__ void __launch_bounds__(256)
gemv_kernel(const float* __restrict__ x, const float* __restrict__ wsum,
            const float* __restrict__ bsum, float* __restrict__ out) {
  __shared__ float sm[256];
  const int b = blockIdx.x;
  const float4* xr = (const float4*)(x + (size_t)b * NF);
  const float4* wr = (const float4*)wsum;

  float acc = 0.0f;
  #pragma unroll 4
  for (int i = threadIdx.x; i < NF / 4; i += 256) {
    float4 xv = xr[i];
    float4 wv = wr[i];
    acc += xv.x * wv.x + xv.y * wv.y + xv.z * wv.z + xv.w * wv.w;
  }
  sm[threadIdx.x] = acc;
  __syncthreads();
  for (int st = 128; st > 0; st >>= 1) {
    if (threadIdx.x < st) sm[threadIdx.x] += sm[threadIdx.x + st];
    __syncthreads();
  }
  if (threadIdx.x == 0) out[b] = sm[0] + bsum[0];
}

// ---------------------------------------------------------------------------
extern "C" void kernel_launch(void* const* d_in, const int* in_sizes, int n_in,
                              void* d_out, int out_size, void* d_ws,
                              size_t ws_size, hipStream_t stream) {
  const float* x    = (const float*)d_in[0];   // (batch, 8192) f32
  const float* W    = (const float*)d_in[1];   // (8192, 8192) f32
  const float* bias = (const float*)d_in[2];   // (8192,) f32
  float* out = (float*)d_out;                  // (batch, 1) f32

  const int batch = in_sizes[0] / NF;          // 2048

  float* ws      = (float*)d_ws;
  float* partial = ws;                          // 16 * 8192 floats
  float* wsum    = ws + (size_t)ROW_CHUNKS * NF; // 8192 floats
  float* bsum    = wsum + NF;                   // 1 float

  dim3 gA(NF / 128, ROW_CHUNKS);               // 64 x 16 blocks, 8 waves each
  colsum_wmma_kernel<<<gA, 256, 0, stream>>>(W, partial);
  reduce_partials_kernel<<<NF / 256, 256, 0, stream>>>(partial, wsum);
  bias_reduce_kernel<<<1, 256, 0, stream>>>(bias, bsum);
  gemv_kernel<<<batch, 256, 0, stream>>>(x, wsum, bsum, out);
}